// MCTSEnhancedTreeReasoningModule_37864431681890
// MI455X (gfx1250) — compile-verified
//
#include <hip/hip_runtime.h>

// ---------------- types ----------------
typedef __bf16  v16bf __attribute__((ext_vector_type(16)));
typedef float   v8f   __attribute__((ext_vector_type(8)));

#define H 768
#define A_DIM 128
#define TOKENS (32 * 2048)
#define M_TILE 32               // tokens per workgroup
#define NBLK (TOKENS / M_TILE)  // 2048 workgroups

// ws bf16 element offsets
#define OFF_PW1  0
#define OFF_PW2  (OFF_PW1 + 768*768)
#define OFF_AE   (OFF_PW2 + 768*128)
#define OFF_TW1  (OFF_AE  + 128*768)
#define OFF_TW2  (OFF_TW1 + 1536*768)
#define OFF_VW1  (OFF_TW2 + 768*768)
#define OFF_OW   (OFF_VW1 + 768*768)

// LDS byte offsets
#define SM_BUF    0          // 32*768 fp32      = 98304
#define SM_XT     98304      // 32*768 bf16      = 49152
#define SM_ACT    147456     // 32*768 bf16      = 49152
#define SM_AEMB   196608     // 32*768 bf16      = 49152
#define SM_RED1   245760     // 32*8 fp32        = 1024
#define SM_RED2   246784     // 32*8 fp32        = 1024
#define SM_PBF    247808     // 32*128 bf16 frag = 8192
#define SM_BYTES  256000

__device__ __forceinline__ unsigned short f2bf(float f) {
    unsigned int u = __float_as_uint(f);
    unsigned int r = (u + 0x7FFFu + ((u >> 16) & 1u)) >> 16;
    return (unsigned short)r;
}
__device__ __forceinline__ float bf2f(unsigned short h) {
    return __uint_as_float(((unsigned int)h) << 16);
}

// WMMA with tied accumulator: D == C register tuple -> no per-iteration copies.
__device__ __forceinline__ void wmma_bf16(v8f& c, v16bf a, v16bf b) {
    asm("v_wmma_f32_16x16x32_bf16 %0, %1, %2, %0" : "+v"(c) : "v"(a), "v"(b));
}
// WMMA(bf16) -> VALU hazard cover (4 co-exec slots per ISA 7.12.1)
__device__ __forceinline__ void wmma_fence() {
    asm volatile("v_nop\n\tv_nop\n\tv_nop\n\tv_nop" ::: );
}

// address (in bf16 elements) of (row, col) inside a 2-Mtile fragment buffer.
// ISA 16-bit A layout: lanes 0-15 hold K{0..7,16..23}, lanes 16-31 hold K{8..15,24..31}
__device__ __forceinline__ int fragAddr(int row, int col) {
    int kt = col >> 5, cc = col & 31, m = row >> 4, r = row & 15;
    int lane = r + ((cc & 8) << 1);
    int idx  = (cc & 7) + ((cc & 16) >> 1);
    return (((kt << 1) + m) << 9) + (lane << 4) + idx;
}

// ---------------- weight packing: fp32 [K,N] row-major -> bf16 WMMA B-fragment order
// B order packed here (contiguous k): lanes 0-15 = col n, K 0..15 ; lanes 16-31 = col n, K 16..31
__global__ void pack_w(const float* __restrict__ W, unsigned short* __restrict__ dst,
                       int K, int N) {
    int d = blockIdx.x * 256 + threadIdx.x;
    int fragId = d >> 9;
    int within = d & 511;
    int lane = within >> 4;
    int i    = within & 15;
    int numKt = K >> 5;
    int nt = fragId / numKt;
    int kt = fragId - nt * numKt;
    int k = (kt << 5) + (lane & 16) + i;
    int n = (nt << 4) + (lane & 15);
    dst[d] = f2bf(W[(size_t)k * N + n]);
}

// ---------------- GEMM core: A (LDS frags, 2 M-tiles) x B (global packed) ----------------
template <int NT>
__device__ __forceinline__ void gemm_acc(const unsigned short* __restrict__ aFrag,
                                         const unsigned short* __restrict__ Bg,
                                         int numKtB, int ktB0, int iters,
                                         int nt0, int lane, v8f (&acc)[2][NT]) {
    for (int kt = 0; kt < iters; ++kt) {
        v16bf a0 = *(const v16bf*)(aFrag + (((kt << 1) + 0) << 9) + (lane << 4));
        v16bf a1 = *(const v16bf*)(aFrag + (((kt << 1) + 1) << 9) + (lane << 4));
        if (kt + 1 < iters)  // pull next K-tile of B toward the WGP (global_prefetch)
            __builtin_prefetch(Bg + (((size_t)(nt0 * numKtB) + ktB0 + kt + 1) << 9) + (lane << 4), 0, 0);
#pragma unroll
        for (int j = 0; j < NT; ++j) {
            const unsigned short* bp =
                Bg + ((((size_t)(nt0 + j) * numKtB) + ktB0 + kt) << 9) + (lane << 4);
            v16bf b = *(const v16bf*)bp;
            wmma_bf16(acc[0][j], a0, b);
            wmma_bf16(acc[1][j], a1, b);
        }
    }
    wmma_fence();
}

// store acc (+bias) as fp32 into an LDS [32 x ncols] buffer
template <int NT>
__device__ __forceinline__ void epi_f32(v8f (&acc)[2][NT], const float* __restrict__ bias,
                                        float* __restrict__ buf, int nt0, int lane, int ncols) {
    int cl = lane & 15, hi8 = ((lane >> 4) & 1) << 3;
#pragma unroll
    for (int m = 0; m < 2; ++m)
#pragma unroll
        for (int j = 0; j < NT; ++j) {
            int col = (nt0 + j) * 16 + cl;
            float b = bias[col];
#pragma unroll
            for (int v = 0; v < 8; ++v) {
                int row = m * 16 + v + hi8;
                buf[row * ncols + col] = acc[m][j][v] + b;
            }
        }
}

// LayerNorm (+optional exact GELU) from fp32 LDS buffer -> bf16 fragment buffer
__device__ __forceinline__ void ln_act(const float* __restrict__ buf,
                                       const float* __restrict__ g, const float* __restrict__ b,
                                       unsigned short* __restrict__ dst,
                                       float* red1, float* red2, int tid, bool gelu) {
    int r = tid >> 3, ch = tid & 7, c0 = ch * 96;
    const float* rowp = buf + r * H;
    float s = 0.f, s2 = 0.f;
    for (int i = 0; i < 96; ++i) { float x = rowp[c0 + i]; s += x; s2 += x * x; }
    red1[r * 8 + ch] = s; red2[r * 8 + ch] = s2;
    __syncthreads();
    float S = 0.f, S2 = 0.f;
    for (int i = 0; i < 8; ++i) { S += red1[r * 8 + i]; S2 += red2[r * 8 + i]; }
    float mean = S * (1.0f / H);
    float var  = S2 * (1.0f / H) - mean * mean;
    float inv  = rsqrtf(var + 1e-5f);
    for (int i = 0; i < 96; ++i) {
        int c = c0 + i;
        float x = (rowp[c] - mean) * inv * g[c] + b[c];
        if (gelu) x = 0.5f * x * (1.0f + erff(x * 0.70710678118f));
        dst[fragAddr(r, c)] = f2bf(x);
    }
    __syncthreads();
}

struct Params {
    const float* X;
    const unsigned short *pW1f, *pW2f, *aEf, *tW1f, *tW2f, *vW1f, *oWf;
    const float *pb1, *pg, *pB, *pb2;
    const float *tb1, *tg, *tB, *tb2;
    const float *vb1, *vg, *vB, *vW2, *vb2;
    const float *og, *oB, *ob;
    float *out, *probs, *reward;
};

__global__ __launch_bounds__(256, 1)
void mcts_fused_kernel(Params P) {
    extern __shared__ char smem[];
    float*          buf   = (float*)(smem + SM_BUF);
    unsigned short* xt    = (unsigned short*)(smem + SM_XT);
    unsigned short* act   = (unsigned short*)(smem + SM_ACT);
    unsigned short* aemb  = (unsigned short*)(smem + SM_AEMB);
    float*          red1  = (float*)(smem + SM_RED1);
    float*          red2  = (float*)(smem + SM_RED2);
    unsigned short* pbf   = (unsigned short*)(smem + SM_PBF);

    const int tid    = threadIdx.x;
    const int lane   = tid & 31;
    const int wave   = tid >> 5;
    const int token0 = blockIdx.x * M_TILE;
    const int nt6    = wave * 6;
    const int cl  = lane & 15;
    const int hi8 = ((lane >> 4) & 1) << 3;

    // ---- load x tile -> bf16 A fragments ----
    {
        int r = tid >> 3, ch = tid & 7, c0 = ch * 96;
        const float* xr = P.X + (size_t)(token0 + r) * H;
        for (int i = 0; i < 96; ++i) {
            int c = c0 + i;
            xt[fragAddr(r, c)] = f2bf(xr[c]);
        }
    }
    __syncthreads();

    // ---- Stage 1: h_pre = x @ pW1 + pb1 ----
    {
        v8f acc[2][6];
        for (int m = 0; m < 2; ++m) for (int j = 0; j < 6; ++j) for (int v = 0; v < 8; ++v) acc[m][j][v] = 0.f;
        gemm_acc<6>(xt, P.pW1f, 24, 0, 24, nt6, lane, acc);
        epi_f32<6>(acc, P.pb1, buf, nt6, lane, H);
    }
    __syncthreads();
    ln_act(buf, P.pg, P.pB, act, red1, red2, tid, true);   // h -> act frags

    // ---- Stage 2: logits = h @ pW2 + pb2 ----
    {
        v8f acc[2][1];
        for (int m = 0; m < 2; ++m) for (int v = 0; v < 8; ++v) acc[m][0][v] = 0.f;
        gemm_acc<1>(act, P.pW2f, 24, 0, 24, wave, lane, acc);
        epi_f32<1>(acc, P.pb2, buf, wave, lane, A_DIM);    // buf now [32 x 128] logits
    }
    __syncthreads();

    // ---- softmax -> probs (global fp32) + probs bf16 frags ----
    {
        int r = tid >> 3, ch = tid & 7, c0 = ch * 16;
        float* lrow = buf + r * A_DIM;
        float mx = -3.0e38f;
        for (int i = 0; i < 16; ++i) mx = fmaxf(mx, lrow[c0 + i]);
        red1[r * 8 + ch] = mx;
        __syncthreads();
        float M = -3.0e38f;
        for (int i = 0; i < 8; ++i) M = fmaxf(M, red1[r * 8 + i]);
        float s = 0.f;
        for (int i = 0; i < 16; ++i) { float e = __expf(lrow[c0 + i] - M); lrow[c0 + i] = e; s += e; }
        red2[r * 8 + ch] = s;
        __syncthreads();
        float S = 0.f;
        for (int i = 0; i < 8; ++i) S += red2[r * 8 + i];
        float inv = 1.0f / S;
        float* prow = P.probs + (size_t)(token0 + r) * A_DIM;
        for (int i = 0; i < 16; ++i) {
            int c = c0 + i;
            float p = lrow[c] * inv;
            prow[c] = p;
            pbf[fragAddr(r, c)] = f2bf(p);
        }
    }
    __syncthreads();

    // ---- Stage 3: a_emb = probs @ action_emb ----
    {
        v8f acc[2][6];
        for (int m = 0; m < 2; ++m) for (int j = 0; j < 6; ++j) for (int v = 0; v < 8; ++v) acc[m][j][v] = 0.f;
        gemm_acc<6>(pbf, P.aEf, 4, 0, 4, nt6, lane, acc);
#pragma unroll
        for (int m = 0; m < 2; ++m)
#pragma unroll
            for (int j = 0; j < 6; ++j) {
                int col = (nt6 + j) * 16 + cl;
#pragma unroll
                for (int v = 0; v < 8; ++v) {
                    int row = m * 16 + v + hi8;
                    aemb[fragAddr(row, col)] = f2bf(acc[m][j][v]);
                }
            }
    }
    __syncthreads();

    // ---- Stage 4: t_pre = cat(x, a_emb) @ tW1 + tb1 ----
    {
        v8f acc[2][6];
        for (int m = 0; m < 2; ++m) for (int j = 0; j < 6; ++j) for (int v = 0; v < 8; ++v) acc[m][j][v] = 0.f;
        gemm_acc<6>(xt,   P.tW1f, 48, 0,  24, nt6, lane, acc);
        gemm_acc<6>(aemb, P.tW1f, 48, 24, 24, nt6, lane, acc);
        epi_f32<6>(acc, P.tb1, buf, nt6, lane, H);
    }
    __syncthreads();
    ln_act(buf, P.tg, P.tB, act, red1, red2, tid, true);   // t -> act frags

    // ---- Stage 5: next = t @ tW2 + tb2 + x ----
    {
        v8f acc[2][6];
        for (int m = 0; m < 2; ++m) for (int j = 0; j < 6; ++j) for (int v = 0; v < 8; ++v) acc[m][j][v] = 0.f;
        gemm_acc<6>(act, P.tW2f, 24, 0, 24, nt6, lane, acc);
#pragma unroll
        for (int m = 0; m < 2; ++m)
#pragma unroll
            for (int j = 0; j < 6; ++j) {
                int col = (nt6 + j) * 16 + cl;
                float bb = P.tb2[col];
#pragma unroll
                for (int v = 0; v < 8; ++v) {
                    int row = m * 16 + v + hi8;
                    float val = acc[m][j][v] + bb + P.X[(size_t)(token0 + row) * H + col];
                    buf[row * H + col] = val;                 // fp32 next_state
                    aemb[fragAddr(row, col)] = f2bf(val);     // bf16 frags (value head input)
                }
            }
    }
    __syncthreads();

    // ---- Stage 8 (reordered): ln(next) -> act frags ----
    ln_act(buf, P.og, P.oB, act, red1, red2, tid, false);

    // ---- Stage 9: out = ln(next) @ oW + ob  (write straight to HBM) ----
    {
        v8f acc[2][6];
        for (int m = 0; m < 2; ++m) for (int j = 0; j < 6; ++j) for (int v = 0; v < 8; ++v) acc[m][j][v] = 0.f;
        gemm_acc<6>(act, P.oWf, 24, 0, 24, nt6, lane, acc);
#pragma unroll
        for (int m = 0; m < 2; ++m)
#pragma unroll
            for (int j = 0; j < 6; ++j) {
                int col = (nt6 + j) * 16 + cl;
                float bb = P.ob[col];
#pragma unroll
                for (int v = 0; v < 8; ++v) {
                    int row = m * 16 + v + hi8;
                    P.out[(size_t)(token0 + row) * H + col] = acc[m][j][v] + bb;
                }
            }
    }

    // ---- Stage 6: v_pre = next @ vW1 + vb1 ----
    {
        v8f acc[2][6];
        for (int m = 0; m < 2; ++m) for (int j = 0; j < 6; ++j) for (int v = 0; v < 8; ++v) acc[m][j][v] = 0.f;
        gemm_acc<6>(aemb, P.vW1f, 24, 0, 24, nt6, lane, acc);
        __syncthreads();                       // everyone done reading buf (stage-8 LN)
        epi_f32<6>(acc, P.vb1, buf, nt6, lane, H);
    }
    __syncthreads();
    ln_act(buf, P.vg, P.vB, act, red1, red2, tid, true);   // v -> act frags

    // ---- Stage 7: value/reward ----
    {
        int r = tid >> 3, ch = tid & 7, c0 = ch * 96;
        float s = 0.f;
        for (int i = 0; i < 96; ++i) {
            int c = c0 + i;
            s += bf2f(act[fragAddr(r, c)]) * P.vW2[c];
        }
        red1[r * 8 + ch] = s;
        __syncthreads();
        if (ch == 0) {
            float S = 0.f;
            for (int i = 0; i < 8; ++i) S += red1[r * 8 + i];
            float value = tanhf(S + P.vb2[0]) * 0.1f;
            P.reward[token0 + r] = value * 0.1f;
        }
    }
}

extern "C" void kernel_launch(void* const* d_in, const int* in_sizes, int n_in,
                              void* d_out, int out_size, void* d_ws, size_t ws_size,
                              hipStream_t stream) {
    const float* X    = (const float*)d_in[0];
    const float* pW1  = (const float*)d_in[1];
    const float* pb1  = (const float*)d_in[2];
    const float* pg   = (const float*)d_in[3];
    const float* pB   = (const float*)d_in[4];
    const float* pW2  = (const float*)d_in[5];
    const float* pb2  = (const float*)d_in[6];
    const float* aE   = (const float*)d_in[7];
    const float* tW1  = (const float*)d_in[8];
    const float* tb1  = (const float*)d_in[9];
    const float* tg   = (const float*)d_in[10];
    const float* tB   = (const float*)d_in[11];
    const float* tW2  = (const float*)d_in[12];
    const float* tb2  = (const float*)d_in[13];
    const float* vW1  = (const float*)d_in[14];
    const float* vb1  = (const float*)d_in[15];
    const float* vg   = (const float*)d_in[16];
    const float* vB   = (const float*)d_in[17];
    const float* vW2  = (const float*)d_in[18];
    const float* vb2  = (const float*)d_in[19];
    const float* og   = (const float*)d_in[20];
    const float* oB   = (const float*)d_in[21];
    const float* oW   = (const float*)d_in[22];
    const float* ob   = (const float*)d_in[23];

    unsigned short* ws = (unsigned short*)d_ws;

    // pack weights fp32 -> bf16 fragment order (runs every launch; deterministic)
    pack_w<<<(768 * 768) / 256, 256, 0, stream>>>(pW1, ws + OFF_PW1, 768, 768);
    pack_w<<<(768 * 128) / 256, 256, 0, stream>>>(pW2, ws + OFF_PW2, 768, 128);
    pack_w<<<(128 * 768) / 256, 256, 0, stream>>>(aE,  ws + OFF_AE,  128, 768);
    pack_w<<<(1536 * 768) / 256, 256, 0, stream>>>(tW1, ws + OFF_TW1, 1536, 768);
    pack_w<<<(768 * 768) / 256, 256, 0, stream>>>(tW2, ws + OFF_TW2, 768, 768);
    pack_w<<<(768 * 768) / 256, 256, 0, stream>>>(vW1, ws + OFF_VW1, 768, 768);
    pack_w<<<(768 * 768) / 256, 256, 0, stream>>>(oW,  ws + OFF_OW,  768, 768);

    float* out_f  = (float*)d_out;
    Params P;
    P.X = X;
    P.pW1f = ws + OFF_PW1; P.pW2f = ws + OFF_PW2; P.aEf = ws + OFF_AE;
    P.tW1f = ws + OFF_TW1; P.tW2f = ws + OFF_TW2; P.vW1f = ws + OFF_VW1; P.oWf = ws + OFF_OW;
    P.pb1 = pb1; P.pg = pg; P.pB = pB; P.pb2 = pb2;
    P.tb1 = tb1; P.tg = tg; P.tB = tB; P.tb2 = tb2;
    P.vb1 = vb1; P.vg = vg; P.vB = vB; P.vW2 = vW2; P.vb2 = vb2;
    P.og = og; P.oB = oB; P.ob = ob;
    P.out    = out_f;
    P.probs  = out_f + (size_t)TOKENS * H;
    P.reward = out_f + (size_t)TOKENS * H + (size_t)TOKENS * A_DIM;

    mcts_fused_kernel<<<NBLK, 256, SM_BYTES, stream>>>(P);
}